// S4_base_85813446574301
// MI455X (gfx1250) — compile-verified
//
#include <hip/hip_runtime.h>
#include <hip/hip_bf16.h>
#include <math.h>

// S4 forward on MI455X (gfx1250), recast from a 16384-step serial scan into
// kernel-generation via fp32 WMMA GEMMs + a cheap causal convolution.
//   P = inv(I - (d/2)A)   (Newton-Schulz, 4 iters, 8 GEMMs)
//   dA = 2P - I ; dB = d * P @ B
//   kappa[i+128j] = (c^T dA^i) . (dA^{128j} dB)  via U(128x1024) @ V(1024x128)
//   y = kappa (causal*) X + D*X
// Workspace: 8 x 4MB matrices + U/V/K2D/kappa tails  (~33.2 MB required).

typedef float v2f __attribute__((ext_vector_type(2)));
typedef float v8f __attribute__((ext_vector_type(8)));

#define NDIM 1024
#define LSEQ 16384

// ---------------- A1 = I - (d/2)A ;  P0 = I + (d/2)A ----------------------
__global__ void k_build(const float* __restrict__ A, const float* __restrict__ logd,
                        float* __restrict__ A1, float* __restrict__ P0) {
  int idx = blockIdx.x * 256 + threadIdx.x;
  int i = idx >> 10, j = idx & 1023;
  float d = expf(logd[0]);
  float e = 0.5f * d * A[idx];
  float I = (i == j) ? 1.f : 0.f;
  A1[idx] = I - e;
  P0[idx] = I + e;
}

// ---------------- dA = 2P - I ---------------------------------------------
__global__ void k_form_dA(const float* __restrict__ P, float* __restrict__ dA) {
  int idx = blockIdx.x * 256 + threadIdx.x;
  int i = idx >> 10, j = idx & 1023;
  dA[idx] = 2.f * P[idx] - ((i == j) ? 1.f : 0.f);
}

// ---------------- fp32 WMMA GEMM:  D = alpha*(A@B) + beta*Cin -------------
// Block tile 128x128, 4 waves (2x2), wave tile 64x64 (4x4 subtiles),
// K-chunk 32, k-step 4 -> 16 WMMAs per 8 fragment loads.
#define KC 32
__global__ __launch_bounds__(128)
void k_gemm(const float* __restrict__ A, int lda,
            const float* __restrict__ B, int ldb,
            const float* __restrict__ Cin,
            float* __restrict__ D, int ldc,
            int M, int N, int K, float alpha, float beta) {
  // A-tile row stride 36 floats: multiple of 4 (b128 LDS stores) and
  // gcd(36,64)=4, period 16 -> 16-lane column reads hit 16 distinct banks.
  __shared__ float As[128][36];
  __shared__ float Bs[KC][128];

  const int tid  = threadIdx.x;
  const int lane = tid & 31;
  const int wave = tid >> 5;
  const int l16  = lane & 15;
  const int half = lane >> 4;      // 0: lanes 0-15, 1: lanes 16-31
  const int wm   = wave & 1;       // 2 waves along M
  const int wn   = wave >> 1;      // 2 waves along N
  const int bm0  = blockIdx.y * 128;
  const int bn0  = blockIdx.x * 128;

  const bool full = (bm0 + 128 <= M) && (bn0 + 128 <= N);

  v8f acc[4][4];
#pragma unroll
  for (int i = 0; i < 4; ++i)
#pragma unroll
    for (int j = 0; j < 4; ++j)
      acc[i][j] = v8f{0.f, 0.f, 0.f, 0.f, 0.f, 0.f, 0.f, 0.f};

  for (int kc = 0; kc < K; kc += KC) {
    if (full) {
      // vectorized float4 loads (all big GEMMs take this path)
#pragma unroll
      for (int n = 0; n < 8; ++n) {          // A: 128x32 = 1024 float4
        int fidx = tid + n * 128;
        int r = fidx >> 3, c4 = (fidx & 7) * 4;
        *(float4*)&As[r][c4] =
            *(const float4*)&A[(size_t)(bm0 + r) * lda + kc + c4];
      }
#pragma unroll
      for (int n = 0; n < 8; ++n) {          // B: 32x128 = 1024 float4
        int fidx = tid + n * 128;
        int r = fidx >> 5, c4 = (fidx & 31) * 4;
        *(float4*)&Bs[r][c4] =
            *(const float4*)&B[(size_t)(kc + r) * ldb + bn0 + c4];
      }
    } else {
      for (int idx = tid; idx < 128 * KC; idx += 128) {
        int r = idx >> 5, c = idx & 31;
        int gr = bm0 + r;
        As[r][c] = (gr < M) ? A[(size_t)gr * lda + kc + c] : 0.f;
      }
      for (int idx = tid; idx < KC * 128; idx += 128) {
        int r = idx >> 7, c = idx & 127;
        int gc = bn0 + c;
        Bs[r][c] = (gc < N) ? B[(size_t)(kc + r) * ldb + gc] : 0.f;
      }
    }
    __syncthreads();

#pragma unroll
    for (int kk = 0; kk < KC; kk += 4) {
      v2f afr[4], bfr[4];
#pragma unroll
      for (int i = 0; i < 4; ++i) {
        int rr = wm * 64 + i * 16 + l16;
        // A 16x4 layout: VGPR0 = K(0|2), VGPR1 = K(1|3) per lane half
        afr[i].x = As[rr][kk + 2 * half];
        afr[i].y = As[rr][kk + 2 * half + 1];
      }
#pragma unroll
      for (int j = 0; j < 4; ++j) {
        int cc = wn * 64 + j * 16 + l16;
        // B 4x16 layout: rows striped across lanes within a VGPR
        bfr[j].x = Bs[kk + 2 * half][cc];
        bfr[j].y = Bs[kk + 2 * half + 1][cc];
      }
#pragma unroll
      for (int i = 0; i < 4; ++i)
#pragma unroll
        for (int j = 0; j < 4; ++j)
          acc[i][j] = __builtin_amdgcn_wmma_f32_16x16x4_f32(
              false, afr[i], false, bfr[j], (short)0, acc[i][j], false, false);
    }
    __syncthreads();
  }

#pragma unroll
  for (int i = 0; i < 4; ++i)
#pragma unroll
    for (int j = 0; j < 4; ++j) {
      int baseR = bm0 + wm * 64 + i * 16 + 8 * half;  // C/D: VGPR r -> M=r / r+8
      int col   = bn0 + wn * 64 + j * 16 + l16;
#pragma unroll
      for (int r = 0; r < 8; ++r) {
        int row = baseR + r;
        if (row < M && col < N) {
          float v = alpha * acc[i][j][r];
          if (beta != 0.f) v += beta * Cin[(size_t)row * ldc + col];
          D[(size_t)row * ldc + col] = v;
        }
      }
    }
}

// ---------------- y = scale * (M @ x), strided x / y ----------------------
__global__ __launch_bounds__(256)
void k_matvec(const float* __restrict__ Mm, const float* __restrict__ x, int xs,
              float* __restrict__ y, int ys, const float* __restrict__ logd) {
  int lane = threadIdx.x & 31;
  int row  = blockIdx.x * 8 + (threadIdx.x >> 5);
  float acc = 0.f;
  for (int c = lane; c < NDIM; c += 32)
    acc += Mm[(size_t)row * NDIM + c] * x[(size_t)c * xs];
  for (int off = 16; off; off >>= 1)
    acc += __shfl_xor(acc, off, 32);
  if (lane == 0) {
    float s = logd ? expf(logd[0]) : 1.f;
    y[(size_t)row * ys] = s * acc;
  }
}

// ---------------- y = M^T @ x  (coalesced over columns) -------------------
__global__ __launch_bounds__(256)
void k_matvecT(const float* __restrict__ Mm, const float* __restrict__ x,
               float* __restrict__ y) {
  __shared__ float xsm[NDIM];
  int tid = threadIdx.x;
  for (int i = tid; i < NDIM; i += 256) xsm[i] = x[i];
  __syncthreads();
  int col = blockIdx.x * 256 + tid;
  float acc = 0.f;
  for (int i = 0; i < NDIM; ++i)
    acc += Mm[(size_t)i * NDIM + col] * xsm[i];
  y[col] = acc;
}

// kappa_flat[i + 128*j] = K2D[i][j]
__global__ void k_reindex(const float* __restrict__ K2D, float* __restrict__ kap) {
  int m = blockIdx.x * 256 + threadIdx.x;
  kap[m] = K2D[(m & 127) * 128 + (m >> 7)];
}

__global__ void k_conv_init(const float* __restrict__ X,
                            const float* __restrict__ Dsk, float* __restrict__ y) {
  int k = blockIdx.x * 256 + threadIdx.x;
  y[k] = Dsk[0] * X[k];
}

// y[k] += sum_{m<=k} kappa[m] * X[k-m]; m-range split over gridDim.y segments
__global__ __launch_bounds__(256)
void k_conv_acc(const float* __restrict__ X, const float* __restrict__ kap,
                float* __restrict__ y) {
  __shared__ float sk[256];
  __shared__ float sx[512];
  const int tid = threadIdx.x;
  const int kb  = blockIdx.x;   // output tile
  const int seg = blockIdx.y;   // m-tile segment
  const int k0  = kb * 256;
  const int k   = k0 + tid;
  float acc = 0.f;
  bool any = false;
  for (int t = seg; t <= kb; t += 8) {
    int mb = t * 256;
    __syncthreads();
    sk[tid] = kap[mb + tid];
    int base = k0 - mb - 255;
    int i0 = base + tid;
    sx[tid] = (i0 >= 0) ? X[i0] : 0.f;
    int i1 = base + 256 + tid;
    sx[256 + tid] = (i1 >= 0 && i1 < LSEQ) ? X[i1] : 0.f;
    __syncthreads();
    if (t < kb) {
#pragma unroll 4
      for (int mm = 0; mm < 256; ++mm)
        acc += sk[mm] * sx[tid + 255 - mm];
    } else {
      for (int mm = 0; mm <= tid; ++mm)   // diagonal tile: m <= k
        acc += sk[mm] * sx[tid + 255 - mm];
    }
    any = true;
  }
  if (any) atomicAdd(&y[k], acc);
}

// ---------------------------------------------------------------------------
static void launch_gemm(const float* A, int lda, const float* B, int ldb,
                        const float* Cin, float* D, int ldc,
                        int M, int N, int K, float alpha, float beta,
                        hipStream_t stream) {
  dim3 grid((N + 127) / 128, (M + 127) / 128);
  k_gemm<<<grid, 128, 0, stream>>>(A, lda, B, ldb, Cin, D, ldc, M, N, K, alpha, beta);
}

extern "C" void kernel_launch(void* const* d_in, const int* in_sizes, int n_in,
                              void* d_out, int out_size, void* d_ws, size_t ws_size,
                              hipStream_t stream) {
  (void)in_sizes; (void)n_in; (void)out_size; (void)ws_size;
  const float* X    = (const float*)d_in[0];
  const float* A    = (const float*)d_in[1];
  const float* B    = (const float*)d_in[2];
  const float* C    = (const float*)d_in[3];
  const float* Dsk  = (const float*)d_in[4];
  const float* logd = (const float*)d_in[5];
  float* y = (float*)d_out;

  const size_t MSZ = (size_t)NDIM * NDIM;   // 1M floats = 4MB
  float* ws    = (float*)d_ws;
  float* A1    = ws + 0 * MSZ;
  float* P0    = ws + 1 * MSZ;
  float* P1    = ws + 2 * MSZ;
  float* T     = ws + 3 * MSZ;
  float* dA    = ws + 4 * MSZ;
  float* M16   = ws + 5 * MSZ;
  float* M128  = ws + 6 * MSZ;
  float* M2048 = ws + 7 * MSZ;
  float* U     = ws + 8 * MSZ;        // 128 x 1024
  float* V     = U + 128 * 1024;      // 1024 x 128 (ld = 128)
  float* K2D   = V + 1024 * 128;      // 128 x 128
  float* KAP   = K2D + 128 * 128;     // 16384

  // A1 = I - E, P0 = I + E
  k_build<<<4096, 256, 0, stream>>>(A, logd, A1, P0);

  // Newton-Schulz: P <- P(2I - A1 P); residual (E^2)^(2^k)
  float* P = P0; float* Pn = P1;
  for (int it = 0; it < 4; ++it) {
    launch_gemm(A1, NDIM, P, NDIM, nullptr, T, NDIM, NDIM, NDIM, NDIM, 1.f, 0.f, stream);
    launch_gemm(P, NDIM, T, NDIM, P, Pn, NDIM, NDIM, NDIM, NDIM, -1.f, 2.f, stream);
    float* tmp = P; P = Pn; Pn = tmp;
  }
  // after 4 swaps P == P0

  k_form_dA<<<4096, 256, 0, stream>>>(P, dA);
  // dB = d * (P @ B) written into V column 0 (stride 128)
  k_matvec<<<128, 256, 0, stream>>>(P, B, 1, V, 128, logd);

  // Squaring chain: dA^2 .. dA^2048, keeping dA^16 / dA^128 / dA^2048
  float* t1 = A1; float* t2 = T;   // dead after Newton
  launch_gemm(dA,  NDIM, dA,  NDIM, nullptr, t1,    NDIM, NDIM, NDIM, NDIM, 1.f, 0.f, stream); // 2
  launch_gemm(t1,  NDIM, t1,  NDIM, nullptr, t2,    NDIM, NDIM, NDIM, NDIM, 1.f, 0.f, stream); // 4
  launch_gemm(t2,  NDIM, t2,  NDIM, nullptr, t1,    NDIM, NDIM, NDIM, NDIM, 1.f, 0.f, stream); // 8
  launch_gemm(t1,  NDIM, t1,  NDIM, nullptr, M16,   NDIM, NDIM, NDIM, NDIM, 1.f, 0.f, stream); // 16
  launch_gemm(M16, NDIM, M16, NDIM, nullptr, t2,    NDIM, NDIM, NDIM, NDIM, 1.f, 0.f, stream); // 32
  launch_gemm(t2,  NDIM, t2,  NDIM, nullptr, t1,    NDIM, NDIM, NDIM, NDIM, 1.f, 0.f, stream); // 64
  launch_gemm(t1,  NDIM, t1,  NDIM, nullptr, M128,  NDIM, NDIM, NDIM, NDIM, 1.f, 0.f, stream); // 128
  launch_gemm(M128,NDIM, M128,NDIM, nullptr, t1,    NDIM, NDIM, NDIM, NDIM, 1.f, 0.f, stream); // 256
  launch_gemm(t1,  NDIM, t1,  NDIM, nullptr, t2,    NDIM, NDIM, NDIM, NDIM, 1.f, 0.f, stream); // 512
  launch_gemm(t2,  NDIM, t2,  NDIM, nullptr, t1,    NDIM, NDIM, NDIM, NDIM, 1.f, 0.f, stream); // 1024
  launch_gemm(t1,  NDIM, t1,  NDIM, nullptr, M2048, NDIM, NDIM, NDIM, NDIM, 1.f, 0.f, stream); // 2048

  // U rows: u_0 = c; u_{i+1} = dA^T u_i (16 rows), then UB_{j+1} = UB_j @ M16
  hipMemcpyAsync(U, C, NDIM * sizeof(float), hipMemcpyDeviceToDevice, stream);
  for (int i = 0; i < 15; ++i)
    k_matvecT<<<4, 256, 0, stream>>>(dA, U + i * NDIM, U + (i + 1) * NDIM);
  for (int j = 0; j < 7; ++j)
    launch_gemm(U + 16 * j * NDIM, NDIM, M16, NDIM, nullptr,
                U + 16 * (j + 1) * NDIM, NDIM, 16, NDIM, NDIM, 1.f, 0.f, stream);

  // V cols: v_0 = dB; v_{j+1} = M128 v_j (16 cols), then VB_{k+1} = M2048 @ VB_k
  for (int j = 0; j < 15; ++j)
    k_matvec<<<128, 256, 0, stream>>>(M128, V + j, 128, V + j + 1, 128, nullptr);
  for (int kk = 0; kk < 7; ++kk)
    launch_gemm(M2048, NDIM, V + 16 * kk, 128, nullptr,
                V + 16 * (kk + 1), 128, NDIM, 16, NDIM, 1.f, 0.f, stream);

  // kappa = U @ V, reindex to flat causal kernel
  launch_gemm(U, NDIM, V, 128, nullptr, K2D, 128, 128, 128, NDIM, 1.f, 0.f, stream);
  k_reindex<<<64, 256, 0, stream>>>(K2D, KAP);

  // y = D*X + kappa (causal conv) X
  k_conv_init<<<64, 256, 0, stream>>>(X, Dsk, y);
  k_conv_acc<<<dim3(64, 8), 256, 0, stream>>>(X, KAP, y);
}